// ModelWrapper_27101243638506
// MI455X (gfx1250) — compile-verified
//
#include <hip/hip_runtime.h>

typedef __attribute__((ext_vector_type(2))) float v2f;
typedef __attribute__((ext_vector_type(8))) float v8f;

// One wave computes 16 outputs of one sample's layer via V_WMMA_F32_16X16X4_F32.
// out[b,o] = act( sum_i W[b,i,o] * x[b,i] + bias[b,o] )
// A (16x4) = x[i0..i0+3] broadcast into all 16 rows; B (4x16) = W tile.
// Every row of D is identical => D row0 (acc VGPR0, lanes 0-15) = the 16 outputs.
template <int I, bool RELU>
__global__ void __launch_bounds__(128, 1)
mlp_layer_wmma(const float* __restrict__ params,
               const float* __restrict__ xin,
               float* __restrict__ xout,
               int O,
               long long w_off, long long b_off, long long P)
{
    __shared__ float xs[I];

    const int tid  = threadIdx.x;
    const int lane = tid & 31;
    const int wv   = tid >> 5;
    const int b    = blockIdx.y;
    const int o0   = (blockIdx.x * 4 + wv) * 16;

    // Cooperative load of this sample's activation vector into LDS.
    const float* __restrict__ xp = xin + (size_t)b * (size_t)I;
    #pragma unroll
    for (int i = tid; i < I; i += 128) xs[i] = xp[i];
    __syncthreads();

    const float* __restrict__ W = params + (size_t)b * (size_t)P + (size_t)w_off;

    const int half = lane >> 4;        // 0: lanes 0-15, 1: lanes 16-31
    const int ncol = lane & 15;        // N index within the 16-wide tile
    // Lane's B-matrix element address:
    //   B VGPR0 holds (K = 0 + 2*half, N = ncol), VGPR1 holds (K = 1 + 2*half, N = ncol)
    const float* __restrict__ wp = W + (size_t)(2 * half) * (size_t)O + (o0 + ncol);
    const size_t rowStep = (size_t)4 * (size_t)O;   // advance 4 K-rows

    const float* __restrict__ xr = &xs[2 * half];   // per-lane x base

    v8f acc0 = {}, acc1 = {}, acc2 = {}, acc3 = {};

    #pragma unroll 4
    for (int i0 = 0; i0 < I; i0 += 8) {
        // K-rows i0 .. i0+3
        v2f A0, B0;
        A0.x = xr[i0];
        A0.y = xr[i0 + 1];
        B0.x = wp[0];
        B0.y = wp[O];
        wp += rowStep;
        acc0 = __builtin_amdgcn_wmma_f32_16x16x4_f32(
            false, A0, false, B0, (short)0, acc0, false, false);

        // K-rows i0+4 .. i0+7 (independent accumulator for ILP)
        v2f A1, B1;
        A1.x = xr[i0 + 4];
        A1.y = xr[i0 + 5];
        B1.x = wp[0];
        B1.y = wp[O];
        wp += rowStep;
        acc1 = __builtin_amdgcn_wmma_f32_16x16x4_f32(
            false, A1, false, B1, (short)0, acc1, false, false);

        // rotate accumulators to break RAW chains across the unroll
        v8f t = acc0; acc0 = acc2; acc2 = t;
        t = acc1; acc1 = acc3; acc3 = t;
    }

    // Epilogue: D row 0 = acc VGPR0 on lanes 0-15 (N = lane).
    if (lane < 16) {
        const float* __restrict__ bias = params + (size_t)b * (size_t)P + (size_t)b_off;
        float r = acc0[0] + acc1[0] + acc2[0] + acc3[0] + bias[o0 + lane];
        if (RELU) r = fmaxf(r, 0.0f);
        xout[(size_t)b * (size_t)O + o0 + lane] = r;
    }
}

extern "C" void kernel_launch(void* const* d_in, const int* in_sizes, int n_in,
                              void* d_out, int out_size, void* d_ws, size_t ws_size,
                              hipStream_t stream) {
    (void)in_sizes; (void)n_in; (void)out_size; (void)ws_size;

    const float* params = (const float*)d_in[0];   // (32, 8393728) f32
    const float* x      = (const float*)d_in[1];   // (32, 1024)    f32
    float*       out    = (float*)d_out;           // (32, 1024)    f32

    const int B = 32;
    float* ws0 = (float*)d_ws;                 // (32, 2048) layer-1 activations
    float* ws1 = ws0 + (size_t)B * 2048;       // (32, 2048) layer-2 activations

    const long long P = 8393728LL;             // per-sample param stride
    // layout per sample: W1[1024*2048] b1[2048] W2[2048*2048] b2[2048] W3[2048*1024] b3[1024]
    const long long w1 = 0LL,       b1 = 2097152LL;
    const long long w2 = 2099200LL, b2 = 6293504LL;
    const long long w3 = 6295552LL, b3 = 8392704LL;

    // Layer 1: 1024 -> 2048, relu
    mlp_layer_wmma<1024, true><<<dim3(2048 / 64, B), 128, 0, stream>>>(
        params, x, ws0, 2048, w1, b1, P);
    // Layer 2: 2048 -> 2048, relu
    mlp_layer_wmma<2048, true><<<dim3(2048 / 64, B), 128, 0, stream>>>(
        params, ws0, ws1, 2048, w2, b2, P);
    // Layer 3: 2048 -> 1024, linear
    mlp_layer_wmma<2048, false><<<dim3(1024 / 64, B), 128, 0, stream>>>(
        params, ws1, out, 1024, w3, b3, P);
}